// InstanceHead_82454782149366
// MI455X (gfx1250) — compile-verified
//
#include <hip/hip_runtime.h>

typedef __attribute__((ext_vector_type(2))) float v2f;
typedef __attribute__((ext_vector_type(8))) float v8f;

#define M_CEN 256
#define D_FEAT 64
#define TILE 16
#define ITERS 8
#define ROWS_PER_BLOCK (TILE * ITERS)

__global__ __launch_bounds__(256)
void instancehead_wmma_kernel(const int* __restrict__ clu_coords,
                              const float* __restrict__ clu_feats,
                              const int* __restrict__ cen_coords,
                              const float* __restrict__ cen_feats,
                              float* __restrict__ out,
                              int n)
{
    __shared__ float s_attn[TILE * M_CEN];   // 16 KB: softmax weights for current row-tile
    __shared__ float s_red[TILE][16];        // 1 KB : per-row reduction scratch
    __shared__ int   s_cc[M_CEN * 4];        // 4 KB : centroid coords (batch,x,y,z)

    const int tid  = threadIdx.x;
    const int lane = tid & 31;
    const int wave = tid >> 5;
    const int ln15 = lane & 15;
    const int lg   = lane >> 4;      // lane group (0: lanes 0-15, 1: lanes 16-31)
    const int koff = lg * 2;         // fp32 WMMA A/B K-pair select per lane group

    // ---- stage centroid coords into LDS ----
    for (int i = tid; i < M_CEN * 4; i += 256)
        s_cc[i] = cen_coords[i];

    // ---- load B fragments (centroid feats, col-tiles 2*wave, 2*wave+1) ----
    // B layout (16x16x4 f32): lane ln15 = column n within tile; VGPR i holds K = 2*lg + i.
    v2f bfrag[2][16];
#pragma unroll
    for (int t = 0; t < 2; ++t) {
        const int c0 = (wave * 2 + t) * TILE;
        const float* bp = cen_feats + (size_t)(c0 + ln15) * D_FEAT + koff;
#pragma unroll
        for (int kk = 0; kk < 16; ++kk)
            bfrag[t][kk] = *(const v2f*)(bp + 4 * kk);
    }
    __syncthreads();

    const int row = tid >> 4;   // 0..15 : local row for softmax phase
    const int grp = tid & 15;   // 0..15 : 16-column group for softmax phase

    for (int it = 0; it < ITERS; ++it) {
        const int r0 = blockIdx.x * ROWS_PER_BLOCK + it * TILE;

        // ================= softmax phase (16 rows x 256 cols) =================
        const int gr = r0 + row;
        int cb = 0, cx = 0, cy = 0, cz = 0;
        if (gr < n) {
            cb = clu_coords[gr * 4 + 0];
            cx = clu_coords[gr * 4 + 1];
            cy = clu_coords[gr * 4 + 2];
            cz = clu_coords[gr * 4 + 3];
        }
        float dloc[16];
        float dmin = 3.4e38f;
#pragma unroll
        for (int j = 0; j < 16; ++j) {
            const int c = grp * 16 + j;
            const bool ok = (gr < n) && (s_cc[c * 4 + 0] == cb);
            const float dx = (float)(cx - s_cc[c * 4 + 1]);
            const float dy = (float)(cy - s_cc[c * 4 + 2]);
            const float dz = (float)(cz - s_cc[c * 4 + 3]);
            float d = __builtin_sqrtf(dx * dx + dy * dy + dz * dz);
            d = fmaxf(d, 0.1f);
            dloc[j] = ok ? d : -1.0f;       // negative marks cross-batch (masked)
            if (ok) dmin = fminf(dmin, d);
        }
        s_red[row][grp] = dmin;
        __syncthreads();
        float rmin = s_red[row][0];
#pragma unroll
        for (int q = 1; q < 16; ++q) rmin = fminf(rmin, s_red[row][q]);
        // max logit = -rmin ; exp(logit - maxlogit) = exp(rmin - dist)
        float lsum = 0.0f;
#pragma unroll
        for (int j = 0; j < 16; ++j) {
            const int c = grp * 16 + j;
            float e;
            if (dloc[j] >= 0.0f) { e = __expf(rmin - dloc[j]); lsum += e; }
            else                 { e = -1.0f; }   // keep masked entries negative
            s_attn[row * M_CEN + c] = e;
        }
        __syncthreads();                 // all rmin reads done before s_red reuse
        s_red[row][grp] = lsum;
        __syncthreads();
        float rsum = 0.0f;
#pragma unroll
        for (int q = 0; q < 16; ++q) rsum += s_red[row][q];
        const float inv = (rsum > 0.0f) ? (1.0f / rsum) : 0.0f;
#pragma unroll
        for (int j = 0; j < 16; ++j)
            s_attn[row * M_CEN + grp * 16 + j] *= inv;  // masked: -inv stays < 0
        __syncthreads();

        // ================= GEMM phase: fp32 WMMA 16x16x4, K = 64 =================
        // A layout: lane ln15 = row m within tile; VGPR i holds K = 2*lg + i.
        const int arow = r0 + ln15;
        v2f afrag[16];
        if (arow < n) {
            const float* ap = clu_feats + (size_t)arow * D_FEAT + koff;
#pragma unroll
            for (int kk = 0; kk < 16; ++kk)
                afrag[kk] = *(const v2f*)(ap + 4 * kk);
        } else {
#pragma unroll
            for (int kk = 0; kk < 16; ++kk)
                afrag[kk] = (v2f)(0.0f);
        }

        // prefetch next row-tile's A rows into cache (global_prefetch_b8)
        if (it + 1 < ITERS) {
            const int pr = r0 + TILE + ln15;
            if (pr < n) __builtin_prefetch(clu_feats + (size_t)pr * D_FEAT, 0, 1);
        }

#pragma unroll
        for (int t = 0; t < 2; ++t) {
            v8f acc = {};
#pragma unroll
            for (int kk = 0; kk < 16; ++kk) {
                acc = __builtin_amdgcn_wmma_f32_16x16x4_f32(
                    /*neg_a=*/false, afrag[kk],
                    /*neg_b=*/false, bfrag[t][kk],
                    /*c_mod=*/(short)0, acc,
                    /*reuse_a=*/false, /*reuse_b=*/false);
            }
            // C/D layout: VGPR j -> M = j + 8*lg, N = ln15 (within tile)
            const int c0   = (wave * 2 + t) * TILE;
            const int ncol = c0 + ln15;
#pragma unroll
            for (int j = 0; j < 8; ++j) {
                const int m  = j + 8 * lg;
                const int g2 = r0 + m;
                if (g2 < n) {
                    const float a = s_attn[m * M_CEN + ncol];
                    float v;
                    if (a < 0.0f) {
                        v = -__builtin_inff();          // cross-batch -> -inf
                    } else {
                        v = acc[j] * a;
                        v = fminf(fmaxf(v, -10.0f), 10.0f);
                    }
                    out[(size_t)g2 * M_CEN + ncol] = v;
                }
            }
        }
        __syncthreads();   // protect s_attn before next iteration overwrites it
    }
}

extern "C" void kernel_launch(void* const* d_in, const int* in_sizes, int n_in,
                              void* d_out, int out_size, void* d_ws, size_t ws_size,
                              hipStream_t stream) {
    (void)n_in; (void)out_size; (void)d_ws; (void)ws_size;
    const int*   clu_coords = (const int*)  d_in[0];
    const float* clu_feats  = (const float*)d_in[1];
    const int*   cen_coords = (const int*)  d_in[2];
    const float* cen_feats  = (const float*)d_in[3];
    float*       out        = (float*)d_out;

    const int n = in_sizes[1] / D_FEAT;   // number of clusters
    const int blocks = (n + ROWS_PER_BLOCK - 1) / ROWS_PER_BLOCK;

    instancehead_wmma_kernel<<<dim3(blocks), dim3(256), 0, stream>>>(
        clu_coords, clu_feats, cen_coords, cen_feats, out, n);
}